// Transformer_81277961109633
// MI455X (gfx1250) — compile-verified
//
#include <hip/hip_runtime.h>

// -------- CDNA5 fp32 WMMA helpers (V_WMMA_F32_16X16X4_F32, wave32) --------
typedef __attribute__((ext_vector_type(2))) float v2f;
typedef __attribute__((ext_vector_type(8))) float v8f;

__device__ __forceinline__ v8f wmma4(v2f a, v2f b, v8f c) {
  // 8 args: (neg_a, A, neg_b, B, c_mod, C, reuse_a, reuse_b)
  return __builtin_amdgcn_wmma_f32_16x16x4_f32(false, a, false, b, (short)0, c,
                                               false, false);
}

__device__ __forceinline__ bool wave_any(bool p) {
#if __has_builtin(__builtin_amdgcn_ballot_w32)
  return __builtin_amdgcn_ballot_w32(p) != 0u;
#else
  return true;
#endif
}

#define CDIM 512
#define NPIX 16384
#define NLAB 8
#define MAT  (512 * 512)
#define ALPHA 0.6f

// ---------------- label counts + validity ----------------
__global__ void count_kernel(const int* __restrict__ segc,
                             const int* __restrict__ segs,
                             float* __restrict__ cnt, float* __restrict__ vldf) {
  __shared__ int sc[16];
  int t = threadIdx.x;
  if (t < 16) sc[t] = 0;
  __syncthreads();
  for (int n = t; n < NPIX; n += 256) {
    atomicAdd(&sc[segc[n] & 7], 1);
    atomicAdd(&sc[8 + (segs[n] & 7)], 1);
  }
  __syncthreads();
  if (t < 16) cnt[t] = (float)sc[t];
  if (t < 8) {
    float nc = (float)sc[t], ns = (float)sc[8 + t];
    vldf[t] = (nc > 10.f && ns > 10.f && nc < 100.f * ns && ns < 100.f * nc)
                  ? 1.f : 0.f;
  }
}

// ---------------- per-label means: mu[(which*8+l)*512 + c] ----------------
__global__ void sums_kernel(const float* __restrict__ Xc,
                            const float* __restrict__ Xs,
                            const int* __restrict__ segc,
                            const int* __restrict__ segs,
                            const float* __restrict__ cnt,
                            float* __restrict__ mu) {
  int c = blockIdx.x;
  int which = blockIdx.y;
  const float* X = which ? Xs : Xc;
  const int* seg = which ? segs : segc;
  int t = threadIdx.x;
  float s[8] = {0, 0, 0, 0, 0, 0, 0, 0};
  for (int n = t; n < NPIX; n += 256) {
    float x = X[c * NPIX + n];
    int lab = seg[n];
#pragma unroll
    for (int l = 0; l < 8; ++l) s[l] += (lab == l) ? x : 0.f;
  }
  __shared__ float red[8 * 256];
#pragma unroll
  for (int l = 0; l < 8; ++l) red[l * 256 + t] = s[l];
  __syncthreads();
  for (int off = 128; off > 0; off >>= 1) {
    if (t < off) {
#pragma unroll
      for (int l = 0; l < 8; ++l) red[l * 256 + t] += red[l * 256 + t + off];
    }
    __syncthreads();
  }
  if (t < 8)
    mu[(which * 8 + t) * CDIM + c] =
        red[t * 256] / fmaxf(cnt[which * 8 + t], 1.f);
}

// --------- masked covariance via WMMA: cov = (Σ m x x^T − n μμ^T)/(n−1) ---------
__global__ __launch_bounds__(32) void cov_kernel(
    const float* __restrict__ X, const int* __restrict__ seg,
    const float* __restrict__ mu, const float* __restrict__ cnt,
    float* __restrict__ covOut, int matBase, int addI) {
  int tM = blockIdx.x, tN = blockIdx.y, l = blockIdx.z;
  int lane = threadIdx.x, r = lane & 15, h = lane >> 4;
  const float* rowA = X + (tM * 16 + r) * NPIX;
  const float* rowB = X + (tN * 16 + r) * NPIX;
  v8f acc = {};
#pragma unroll 4
  for (int k = 0; k < NPIX; k += 4) {
    int kk = k + 2 * h;
    int l0 = seg[kk], l1 = seg[kk + 1];
    v2f xa = *(const v2f*)(rowA + kk);
    v2f a;
    a.x = (l0 == l) ? xa.x : 0.f;
    a.y = (l1 == l) ? xa.y : 0.f;
    v2f b = *(const v2f*)(rowB + kk);
    acc = wmma4(a, b, acc);
  }
  const float* muL = mu + (matBase + l) * CDIM;
  float nl = cnt[matBase + l];
  float inv = 1.f / fmaxf(nl - 1.f, 1.f);
  float mud = muL[tN * 16 + r];
  float* D = covOut + (size_t)(matBase + l) * MAT;
  int d = tN * 16 + r;
#pragma unroll
  for (int v = 0; v < 8; ++v) {
    int c = tM * 16 + v + 8 * h;
    float val = (acc[v] - nl * muL[c] * mud) * inv;
    if (addI && c == d) val += 1.f;
    D[c * CDIM + d] = val;
  }
}

// ---------------- trace of each of the 16 matrices ----------------
__global__ void trace_kernel(const float* __restrict__ covT,
                             float* __restrict__ tr) {
  int m = blockIdx.x, t = threadIdx.x;
  __shared__ float red[64];
  float s = 0.f;
  for (int i = t; i < CDIM; i += 64) s += covT[(size_t)m * MAT + i * CDIM + i];
  red[t] = s;
  __syncthreads();
  for (int off = 32; off > 0; off >>= 1) {
    if (t < off) red[t] += red[t + off];
    __syncthreads();
  }
  if (t == 0) tr[m] = red[0];
}

// ---------------- Y0 = A/tr, Z0 = I ----------------
__global__ void init_yz_kernel(const float* __restrict__ covT,
                               const float* __restrict__ tr,
                               float* __restrict__ Y, float* __restrict__ Z) {
  size_t i = (size_t)blockIdx.x * 256 + threadIdx.x;
  int m = (int)(i >> 18);
  int rem = (int)(i & (MAT - 1));
  int row = rem >> 9, col = rem & 511;
  Y[i] = covT[i] / tr[m];
  Z[i] = (row == col) ? 1.f : 0.f;
}

// ---------------- 512^3 WMMA GEMM tile (one wave / 16x16 tile) ----------------
__device__ __forceinline__ void gemm_tile_512(const float* __restrict__ A,
                                              const float* __restrict__ B,
                                              v8f& acc, int tM, int tN, int r,
                                              int h) {
  const float* rowA = A + (tM * 16 + r) * CDIM;
#pragma unroll 4
  for (int k = 0; k < CDIM; k += 4) {
    int kk = k + 2 * h;
    v2f a = *(const v2f*)(rowA + kk);
    const float* pb = B + kk * CDIM + tN * 16 + r;
    v2f b;
    b.x = pb[0];
    b.y = pb[CDIM];
    acc = wmma4(a, b, acc);
  }
}

// ---------------- Newton–Schulz: T = 1.5 I − 0.5 Z·Y ----------------
__global__ __launch_bounds__(32) void ns_T_kernel(const float* __restrict__ Z,
                                                  const float* __restrict__ Y,
                                                  float* __restrict__ T) {
  int tM = blockIdx.x, tN = blockIdx.y, m = blockIdx.z;
  int lane = threadIdx.x, r = lane & 15, h = lane >> 4;
  v8f acc = {};
  gemm_tile_512(Z + (size_t)m * MAT, Y + (size_t)m * MAT, acc, tM, tN, r, h);
  float* D = T + (size_t)m * MAT;
  int d = tN * 16 + r;
#pragma unroll
  for (int v = 0; v < 8; ++v) {
    int c = tM * 16 + v + 8 * h;
    D[c * CDIM + d] = ((c == d) ? 1.5f : 0.f) - 0.5f * acc[v];
  }
}

// ---------------- Newton–Schulz: Yb = Y·T (z<16), Zb = T·Z (z>=16) ----------------
__global__ __launch_bounds__(32) void ns_update_kernel(
    const float* __restrict__ Y, const float* __restrict__ Z,
    const float* __restrict__ T, float* __restrict__ Yb,
    float* __restrict__ Zb) {
  int tM = blockIdx.x, tN = blockIdx.y, z = blockIdx.z;
  int lane = threadIdx.x, r = lane & 15, h = lane >> 4;
  const float *A, *B;
  float* D;
  if (z < 16) {
    A = Y + (size_t)z * MAT;
    B = T + (size_t)z * MAT;
    D = Yb + (size_t)z * MAT;
  } else {
    int m = z - 16;
    A = T + (size_t)m * MAT;
    B = Z + (size_t)m * MAT;
    D = Zb + (size_t)m * MAT;
  }
  v8f acc = {};
  gemm_tile_512(A, B, acc, tM, tN, r, h);
  int d = tN * 16 + r;
#pragma unroll
  for (int v = 0; v < 8; ++v) D[(tM * 16 + v + 8 * h) * CDIM + d] = acc[v];
}

// ------- M_l = weight·sqrt(tr_s/tr_c) · Y_style[l] · Z_content[l] -------
__global__ __launch_bounds__(32) void compose_kernel(
    const float* __restrict__ Yf, const float* __restrict__ Zf,
    const float* __restrict__ tr, const float* __restrict__ wgt,
    float* __restrict__ Mm) {
  int tM = blockIdx.x, tN = blockIdx.y, l = blockIdx.z;
  int lane = threadIdx.x, r = lane & 15, h = lane >> 4;
  float g = wgt[0] * sqrtf(tr[8 + l] / tr[l]);
  v8f acc = {};
  gemm_tile_512(Yf + (size_t)(8 + l) * MAT, Zf + (size_t)l * MAT, acc, tM, tN,
                r, h);
  float* D = Mm + (size_t)l * MAT;
  int d = tN * 16 + r;
#pragma unroll
  for (int v = 0; v < 8; ++v) D[(tM * 16 + v + 8 * h) * CDIM + d] = g * acc[v];
}

// ---------------- b_l = mu_s[l] − M_l · mu_c[l] ----------------
__global__ void bvec_kernel(const float* __restrict__ Mm,
                            const float* __restrict__ mu,
                            float* __restrict__ bv) {
  int l = blockIdx.x, c = threadIdx.x;
  __shared__ float muc[CDIM];
  muc[c] = mu[l * CDIM + c];
  __syncthreads();
  const float* row = Mm + (size_t)l * MAT + c * CDIM;
  float dot = 0.f;
  for (int d = 0; d < CDIM; ++d) dot += row[d] * muc[d];
  bv[l * CDIM + c] = mu[(8 + l) * CDIM + c] - dot;
}

// ----- apply: out = α(Σ_l m_l(M_l x + b_l) + (1−cov)x) + (1−α)x -----
__global__ __launch_bounds__(32) void apply_kernel(
    const float* __restrict__ Xc, const int* __restrict__ seg,
    const float* __restrict__ Mm, const float* __restrict__ bv,
    const float* __restrict__ vldf, float* __restrict__ out) {
  int tC = blockIdx.x, tN = blockIdx.y;
  int lane = threadIdx.x, r = lane & 15, h = lane >> 4;
  int n = tN * 16 + r;
  int lab = seg[n] & 7;
  float val = vldf[lab];
  v8f acc = {};
  for (int l = 0; l < 8; ++l) {
    bool sel = (lab == l) && (val > 0.5f);
    if (wave_any(sel)) {  // wave-uniform branch: EXEC stays all-ones for WMMA
      float cm = sel ? 1.f : 0.f;
      const float* rowA = Mm + (size_t)l * MAT + (tC * 16 + r) * CDIM;
#pragma unroll 4
      for (int k = 0; k < CDIM; k += 4) {
        int kk = k + 2 * h;
        v2f a = *(const v2f*)(rowA + kk);
        const float* pb = Xc + kk * NPIX + n;
        v2f b;
        b.x = cm * pb[0];
        b.y = cm * pb[NPIX];
        acc = wmma4(a, b, acc);
      }
    }
  }
#pragma unroll
  for (int v = 0; v < 8; ++v) {
    int c = tC * 16 + v + 8 * h;
    float x = Xc[c * NPIX + n];
    float t = acc[v] + (val > 0.5f ? bv[lab * CDIM + c] : x);
    out[c * NPIX + n] = ALPHA * t + (1.f - ALPHA) * x;
  }
}

// =======================================================================
extern "C" void kernel_launch(void* const* d_in, const int* in_sizes, int n_in,
                              void* d_out, int out_size, void* d_ws,
                              size_t ws_size, hipStream_t stream) {
  const float* Xc = (const float*)d_in[0];
  const float* Xs = (const float*)d_in[1];
  const float* wgt = (const float*)d_in[2];
  const int* segc = (const int*)d_in[3];
  const int* segs = (const int*)d_in[4];
  float* outp = (float*)d_out;

  float* W = (float*)d_ws;
  float* cnt = W;           // [16]
  float* vldf = W + 16;     // [8]
  float* tr = W + 24;       // [16]
  float* mu = W + 64;       // [16][512]
  float* bv = W + 64 + 16 * CDIM;  // [8][512]
  float* covT = W + 16384;         // 16*MAT, reused as T
  float* Y = covT + (size_t)16 * MAT;
  float* Z = Y + (size_t)16 * MAT;
  float* Yb = Z + (size_t)16 * MAT;
  float* Zb = Yb + (size_t)16 * MAT;
  float* Mm = Zb + (size_t)16 * MAT;  // 8*MAT

  count_kernel<<<1, 256, 0, stream>>>(segc, segs, cnt, vldf);
  sums_kernel<<<dim3(CDIM, 2), 256, 0, stream>>>(Xc, Xs, segc, segs, cnt, mu);

  cov_kernel<<<dim3(32, 32, 8), 32, 0, stream>>>(Xc, segc, mu, cnt, covT, 0, 1);
  cov_kernel<<<dim3(32, 32, 8), 32, 0, stream>>>(Xs, segs, mu, cnt, covT, 8, 0);

  trace_kernel<<<16, 64, 0, stream>>>(covT, tr);
  init_yz_kernel<<<(16 * MAT) / 256, 256, 0, stream>>>(covT, tr, Y, Z);

  float* T = covT;  // covariances no longer needed
  float *curY = Y, *curZ = Z, *nxtY = Yb, *nxtZ = Zb;
  for (int it = 0; it < 18; ++it) {
    ns_T_kernel<<<dim3(32, 32, 16), 32, 0, stream>>>(curZ, curY, T);
    ns_update_kernel<<<dim3(32, 32, 32), 32, 0, stream>>>(curY, curZ, T, nxtY,
                                                          nxtZ);
    float* t0 = curY; curY = nxtY; nxtY = t0;
    float* t1 = curZ; curZ = nxtZ; nxtZ = t1;
  }

  compose_kernel<<<dim3(32, 32, 8), 32, 0, stream>>>(curY, curZ, tr, wgt, Mm);
  bvec_kernel<<<8, 512, 0, stream>>>(Mm, mu, bv);
  apply_kernel<<<dim3(32, 1024), 32, 0, stream>>>(Xc, segc, Mm, bv, vldf, outp);
}